// SAGEPROTEINS_65377992180266
// MI455X (gfx1250) — compile-verified
//
#include <hip/hip_runtime.h>

#define N_NODES 50000
#define N_EDGES 600000
#define EMB 128

typedef float v2f __attribute__((ext_vector_type(2)));
typedef float v8f __attribute__((ext_vector_type(8)));

__global__ void zero_kernel(float* __restrict__ p, long n) {
  long i = (long)blockIdx.x * blockDim.x + threadIdx.x;
  if (i < n) p[i] = 0.0f;
}

__global__ void deg_kernel(const int* __restrict__ dst, float* __restrict__ deg, int E) {
  int e = blockIdx.x * blockDim.x + threadIdx.x;
  if (e < E) atomicAdd(&deg[dst[e]], 1.0f);
}

__global__ void recip_kernel(float* __restrict__ deg, int n) {
  int i = blockIdx.x * blockDim.x + threadIdx.x;
  if (i < n) {
    float d = deg[i];
    deg[i] = d > 0.0f ? 1.0f / d : 0.0f;
  }
}

// Edge-parallel scatter: each thread handles one float4 chunk of one edge's row.
// For D=128 a wave32 covers exactly one edge (32 lanes x float4 = 128 floats).
template <int D>
__global__ void scatter_kernel(const float* __restrict__ x, const int* __restrict__ src,
                               const int* __restrict__ dst, float* __restrict__ agg, int E) {
  constexpr int D4 = D / 4;
  long t = (long)blockIdx.x * blockDim.x + threadIdx.x;
  long e = t / D4;
  int c = (int)(t - e * D4);
  if (e >= E) return;
  int s = src[e];
  int d = dst[e];
  float4 v = ((const float4*)(x + (size_t)s * D))[c];
  float* o = agg + (size_t)d * D + (size_t)c * 4;
  atomicAdd(o + 0, v.x);
  atomicAdd(o + 1, v.y);
  atomicAdd(o + 2, v.z);
  atomicAdd(o + 3, v.w);
}

// Fused SAGE layer GEMM:
//   out = [relu]( (agg * deg_inv) @ Wl + bias + h @ Wr [+ h] )
// Block = 16-row M tile x 128 cols; 8 waves, each owns a 16x16 N tile.
// fp32 WMMA: V_WMMA_F32_16X16X4_F32, K stepped by 4.
template <int KDIM, bool RELU, bool RES>
__global__ __launch_bounds__(256) void sage_gemm(
    const float* __restrict__ agg, const float* __restrict__ h,
    const float* __restrict__ deg_inv, const float* __restrict__ Wl,
    const float* __restrict__ Wr, const float* __restrict__ bias,
    float* __restrict__ out) {
  constexpr int PITCH = KDIM + 4;  // pad to avoid 16-way LDS bank conflicts
  __shared__ float sAgg[16 * PITCH];
  __shared__ float sH[16 * PITCH];

  const int m_base = blockIdx.x * 16;
  const int tid = threadIdx.x;
  const int lane = tid & 31;
  const int wave = tid >> 5;   // 0..7 -> N tile
  const int n_base = wave * 16;

  // Cooperative, coalesced load of the two 16xK A-tiles (deg_inv folded in).
  for (int i = tid; i < 16 * KDIM; i += 256) {
    int r = i / KDIM;
    int c = i - r * KDIM;
    sAgg[r * PITCH + c] = agg[(size_t)(m_base + r) * KDIM + c] * deg_inv[m_base + r];
    sH[r * PITCH + c]   = h[(size_t)(m_base + r) * KDIM + c];
  }
  __syncthreads();

  // f32 WMMA fragment layout (ISA 7.12.2):
  //   A 16x4: lane L holds (M = L%16, K = 2*(L/16) + vgpr)
  //   B 4x16: lane L holds (N = L%16, K = 2*(L/16) + vgpr)
  //   C/D 16x16: vgpr r, lane L -> (M = r + 8*(L/16), N = L%16)
  const int mrow = lane & 15;
  const int kgrp = (lane >> 4) << 1;
  const int ncol = lane & 15;

  v8f acc = {};
#pragma unroll
  for (int k0 = 0; k0 < KDIM; k0 += 4) {
    v2f a, b;
    a.x = sAgg[mrow * PITCH + k0 + kgrp + 0];
    a.y = sAgg[mrow * PITCH + k0 + kgrp + 1];
    b.x = Wl[(k0 + kgrp + 0) * EMB + n_base + ncol];
    b.y = Wl[(k0 + kgrp + 1) * EMB + n_base + ncol];
    acc = __builtin_amdgcn_wmma_f32_16x16x4_f32(false, a, false, b, (short)0, acc,
                                                false, false);
  }
#pragma unroll
  for (int k0 = 0; k0 < KDIM; k0 += 4) {
    v2f a, b;
    a.x = sH[mrow * PITCH + k0 + kgrp + 0];
    a.y = sH[mrow * PITCH + k0 + kgrp + 1];
    b.x = Wr[(k0 + kgrp + 0) * EMB + n_base + ncol];
    b.y = Wr[(k0 + kgrp + 1) * EMB + n_base + ncol];
    acc = __builtin_amdgcn_wmma_f32_16x16x4_f32(false, a, false, b, (short)0, acc,
                                                false, false);
  }

  const float bv = bias[n_base + ncol];
  const int mtop = (lane >> 4) << 3;
#pragma unroll
  for (int r = 0; r < 8; ++r) {
    const int row_in_tile = mtop + r;
    float v = acc[r] + bv;
    if (RES) v += sH[row_in_tile * PITCH + n_base + ncol];  // residual from LDS
    if (RELU) v = fmaxf(v, 0.0f);
    out[(size_t)(m_base + row_in_tile) * EMB + n_base + ncol] = v;
  }
}

extern "C" void kernel_launch(void* const* d_in, const int* in_sizes, int n_in,
                              void* d_out, int out_size, void* d_ws, size_t ws_size,
                              hipStream_t stream) {
  const float* x   = (const float*)d_in[0];
  const int*   src = (const int*)d_in[1];
  const int*   dst = (const int*)d_in[2];
  const float* W1l = (const float*)d_in[3];
  const float* W1r = (const float*)d_in[4];
  const float* b1  = (const float*)d_in[5];
  const float* W2l = (const float*)d_in[6];
  const float* W2r = (const float*)d_in[7];
  const float* b2  = (const float*)d_in[8];
  const float* W3l = (const float*)d_in[9];
  const float* W3r = (const float*)d_in[10];
  const float* b3  = (const float*)d_in[11];
  float* out = (float*)d_out;

  float* ws  = (float*)d_ws;
  float* deg = ws;                                    // N (becomes deg_inv)
  float* agg = ws + N_NODES;                          // N * EMB
  float* h1  = agg + (size_t)N_NODES * EMB;           // N * EMB
  float* h2  = h1 + (size_t)N_NODES * EMB;            // N * EMB

  auto cdiv = [](long a, long b) { return (int)((a + b - 1) / b); };
  const int MB = N_NODES / 16;  // 50000 / 16 = 3125 exactly

  // Degrees (shared by all layers), then invert in place.
  zero_kernel<<<cdiv(N_NODES, 256), 256, 0, stream>>>(deg, N_NODES);
  deg_kernel<<<cdiv(N_EDGES, 256), 256, 0, stream>>>(dst, deg, N_EDGES);
  recip_kernel<<<cdiv(N_NODES, 256), 256, 0, stream>>>(deg, N_NODES);

  // Layer 1 (K = 8): h1 = relu(mean_agg(x) @ W1l + b1 + x @ W1r)
  zero_kernel<<<cdiv((long)N_NODES * 8, 256), 256, 0, stream>>>(agg, (long)N_NODES * 8);
  scatter_kernel<8><<<cdiv((long)N_EDGES * 2, 256), 256, 0, stream>>>(x, src, dst, agg, N_EDGES);
  sage_gemm<8, true, false><<<MB, 256, 0, stream>>>(agg, x, deg, W1l, W1r, b1, h1);

  // Layer 2 (K = 128): h2 = relu(mean_agg(h1) @ W2l + b2 + h1 @ W2r + h1)
  zero_kernel<<<cdiv((long)N_NODES * EMB, 256), 256, 0, stream>>>(agg, (long)N_NODES * EMB);
  scatter_kernel<EMB><<<cdiv((long)N_EDGES * 32, 256), 256, 0, stream>>>(h1, src, dst, agg, N_EDGES);
  sage_gemm<EMB, true, true><<<MB, 256, 0, stream>>>(agg, h1, deg, W2l, W2r, b2, h2);

  // Layer 3 (K = 128): out = mean_agg(h2) @ W3l + b3 + h2 @ W3r + h2
  zero_kernel<<<cdiv((long)N_NODES * EMB, 256), 256, 0, stream>>>(agg, (long)N_NODES * EMB);
  scatter_kernel<EMB><<<cdiv((long)N_EDGES * 32, 256), 256, 0, stream>>>(h2, src, dst, agg, N_EDGES);
  sage_gemm<EMB, false, true><<<MB, 256, 0, stream>>>(agg, h2, deg, W3l, W3r, b3, out);
}